// ContinousPool1D_17798344474706
// MI455X (gfx1250) — compile-verified
//
#include <hip/hip_runtime.h>

// ContinuousPool1D: y[b,i,d] = sum_{j<=i} s_j*x_j*exp(-a*(t_i-t_j)) + beta*b_i
// Decoupled single-pass scan:
//   per-chunk LOCAL scan (carry-free) with V_WMMA_F32_16X16X4_F32 tril-matmul
//   cumsums, then carry fixup y = h_local + h_in*E + beta*b (linearity).
// Memory traffic = 4 reads + 1 write of 67MB = 335MB  ->  ~14.4us at 23.3TB/s.

#define ALPHA_ 0.5f
#define BETA_  0.1f

constexpr int B_ = 8, L_ = 4096, D_ = 512;
constexpr int BLOCK   = 256;            // 8 waves (wave32)
constexpr int DW      = 128;            // d-columns per block
constexpr int NSTRIP  = (B_ * D_) / DW; // 32 strips
constexpr int NCHUNK  = 32;             // chunks along L (lookback chained)
constexpr int CHUNK_L = L_ / NCHUNK;    // 128 timesteps per chunk
constexpr int TILES   = CHUNK_L / 16;   // 8 tiles per chunk
constexpr int LDSW    = 144;            // staging row stride (words)
constexpr int HS2     = 260;            // (h,E) interleaved row stride (words):
                                        // 8*260 % 64 == 32 -> half-wave b64
                                        // writes land on disjoint bank groups
constexpr int SMEM_FLOATS = 32 * LDSW + CHUNK_L * HS2;   // 37888 -> 151.6 KB

// exp(z) via native base-2 v_exp_f32 with pre-folded log2(e)
#define LOG2E_ 1.44269504088896340736f
#if defined(__has_builtin)
#if __has_builtin(__builtin_amdgcn_exp2f)
#define EXP2N_(z) __builtin_amdgcn_exp2f(z)
#else
#define EXP2N_(z) __expf((z) * 0.69314718055994530942f)
#endif
#else
#define EXP2N_(z) __expf((z) * 0.69314718055994530942f)
#endif

typedef float v2f __attribute__((ext_vector_type(2)));
typedef float v8f __attribute__((ext_vector_type(8)));

__global__ void cp1d_init_flags(int* flags, int n) {
  int i = blockIdx.x * blockDim.x + threadIdx.x;
  if (i < n) flags[i] = 0;
}

__global__ __launch_bounds__(BLOCK)
void cp1d_scan_kernel(const float* __restrict__ x, const float* __restrict__ t,
                      const float* __restrict__ s, const float* __restrict__ bb,
                      float* __restrict__ y, float* __restrict__ carry,
                      int* __restrict__ flags)
{
  extern __shared__ float smem[];
  float* lu = smem;                 // 16 x LDSW staging: u = s*x
  float* lt = smem + 16 * LDSW;     // 16 x LDSW staging: t
  float* he = smem + 32 * LDSW;     // CHUNK_L x HS2 : interleaved (h_local, E)

  const int strip = blockIdx.x;            // 0..31
  const int c     = blockIdx.y;            // chunk; y slowest -> producers first
  const int bidx  = strip >> 2;
  const int d0b   = (strip & 3) * DW;

  const int tid  = threadIdx.x;
  const int w    = tid >> 5;
  const int l    = tid & 31;
  const int half = l >> 4;
  const int n    = l & 15;                 // M for A; N for B/C/D
  const int colB = w * 16 + n;
  const int dcol = d0b + colB;

  const long rowBase = (long)bidx * L_;
  const int  i0c     = c * CHUNK_L;

  // A = tril(ones), four 16x4 f32 K-slices. Lane M = l&15; VGPR0/1 hold
  // K = 2*half, 2*half+1 within each slice.
  v2f Aslice[4];
  #pragma unroll
  for (int k = 0; k < 4; ++k) {
    const int K0 = 4 * k + 2 * half;
    Aslice[k].x = (n >= K0)     ? 1.0f : 0.0f;
    Aslice[k].y = (n >= K0 + 1) ? 1.0f : 0.0f;
  }

  // chunk reference time = time of element just before the chunk (dt_0 = 0)
  float tprev = (c == 0) ? t[rowBase * D_ + dcol]
                         : t[(rowBase + i0c - 1) * D_ + dcol];
  float hloc = 0.0f;   // local (carry-free) state
  float Ech  = 1.0f;   // accumulated decay back to chunk reference

  // ================= local scan: NO dependence on other chunks =================
  for (int tt = 0; tt < TILES; ++tt) {
    const int  i0 = i0c + tt * 16;
    const long g0 = (rowBase + i0) * (long)D_ + d0b;

    #pragma unroll
    for (int k = 0; k < 2; ++k) {          // coalesced b128 loads; u = s*x
      const int  f  = tid + k * 256;
      const int  r  = f >> 5;
      const int  cw = (f & 31) << 2;
      const long g  = g0 + (long)r * D_ + cw;
      const float4 x4 = *(const float4*)&x[g];
      const float4 s4 = *(const float4*)&s[g];
      const float4 t4 = *(const float4*)&t[g];
      float4 u4;
      u4.x = x4.x * s4.x; u4.y = x4.y * s4.y;
      u4.z = x4.z * s4.z; u4.w = x4.w * s4.w;
      *(float4*)&lu[r * LDSW + cw] = u4;
      *(float4*)&lt[r * LDSW + cw] = t4;
    }
    __syncthreads();

    // w_j = u_j*exp(+a*(t_j - tprev)); S = tril(1) @ W via 4 chained WMMAs
    v8f acc = {0.f, 0.f, 0.f, 0.f, 0.f, 0.f, 0.f, 0.f};
    #pragma unroll
    for (int k = 0; k < 4; ++k) {
      const int r0 = 4 * k + 2 * half;
      const int r1 = r0 + 1;
      v2f bv;
      bv.x = lu[r0 * LDSW + colB] *
             EXP2N_((ALPHA_ * LOG2E_) * (lt[r0 * LDSW + colB] - tprev));
      bv.y = lu[r1 * LDSW + colB] *
             EXP2N_((ALPHA_ * LOG2E_) * (lt[r1 * LDSW + colB] - tprev));
      acc = __builtin_amdgcn_wmma_f32_16x16x4_f32(
          false, Aslice[k], false, bv, (short)0, acc, false, false);
    }

    // preload this half-wave's 8 row-times, then the exp chain can pipeline
    float te[8];
    #pragma unroll
    for (int v = 0; v < 8; ++v) te[v] = lt[(v + 8 * half) * LDSW + colB];
    float ev[8];
    #pragma unroll
    for (int v = 0; v < 8; ++v)
      ev[v] = EXP2N_((-ALPHA_ * LOG2E_) * (te[v] - tprev));

    // h_i = e_i*(S_i + hloc); E_i = e_i*Ech.  C/D layout: row = v + 8*half.
    float lastH = 0.f, lastE = 0.f;
    #pragma unroll
    for (int v = 0; v < 8; ++v) {
      const int i = (i0 - i0c) + v + 8 * half;          // row within chunk
      lastH = ev[v] * (acc[v] + hloc);
      lastE = ev[v] * Ech;
      float2 rec; rec.x = lastH; rec.y = lastE;
      *(float2*)&he[i * HS2 + 2 * colB] = rec;          // one ds_store_b64
    }
    const float tlast = lt[15 * LDSW + colB];
    // broadcast row-15 state (held by upper half-wave) to all lanes
    hloc  = __shfl(lastH, n | 16, 32);
    Ech   = __shfl(lastE, n | 16, 32);
    tprev = tlast;
    __syncthreads();   // staging reuse + he complete before fixup
  }

  // ================= lookback: receive carry, publish ours fast ===============
  float hin = 0.0f;
  if (c > 0) {
    const int fprev = strip * NCHUNK + (c - 1);
    while (__hip_atomic_load(&flags[fprev], __ATOMIC_ACQUIRE,
                             __HIP_MEMORY_SCOPE_AGENT) != 16) {
      __builtin_amdgcn_s_sleep(1);
    }
    hin = carry[(long)fprev * DW + colB];
  }
  if (c < NCHUNK - 1 && l < 16) {          // one lane per column publishes
    const int fme = strip * NCHUNK + c;
    carry[(long)fme * DW + colB] = hloc + hin * Ech;   // single FMA on chain
    __hip_atomic_fetch_add(&flags[fme], 1, __ATOMIC_RELEASE,
                           __HIP_MEMORY_SCOPE_AGENT);
  }

  // ================= fixup epilogue: y = h_local + hin*E + beta*b =============
  const int cw = (tid & 31) << 2;          // column (constant per thread)
  float4 hin4 = {0.f, 0.f, 0.f, 0.f};
  if (c > 0) {
    const int fprev = strip * NCHUNK + (c - 1);
    hin4 = *(const float4*)&carry[(long)fprev * DW + cw];
  }
  #pragma unroll 4
  for (int r = 0; r < (CHUNK_L * DW) / (BLOCK * 4); ++r) {   // 16 iters
    const int  row = (tid >> 5) + r * 8;                     // 0..127
    const long g   = (rowBase + i0c + row) * (long)D_ + d0b + cw;
    const float4 bv  = *(const float4*)&bb[g];
    const float4 he0 = *(const float4*)&he[row * HS2 + 2 * cw];     // h0,E0,h1,E1
    const float4 he1 = *(const float4*)&he[row * HS2 + 2 * cw + 4]; // h2,E2,h3,E3
    float4 o;
    o.x = he0.x + hin4.x * he0.y + BETA_ * bv.x;
    o.y = he0.z + hin4.y * he0.w + BETA_ * bv.y;
    o.z = he1.x + hin4.z * he1.y + BETA_ * bv.z;
    o.w = he1.z + hin4.w * he1.w + BETA_ * bv.w;
    *(float4*)&y[g] = o;
  }
}

extern "C" void kernel_launch(void* const* d_in, const int* in_sizes, int n_in,
                              void* d_out, int out_size, void* d_ws, size_t ws_size,
                              hipStream_t stream) {
  (void)in_sizes; (void)n_in; (void)out_size; (void)ws_size;
  const float* x = (const float*)d_in[0];
  const float* t = (const float*)d_in[1];
  const float* s = (const float*)d_in[2];
  const float* b = (const float*)d_in[3];
  float* y = (float*)d_out;

  int*   flags = (int*)d_ws;                      // NSTRIP*NCHUNK = 1024 ints
  float* carry = (float*)((char*)d_ws + 8192);    // 1024*128 floats = 512 KB

  hipLaunchKernelGGL(cp1d_init_flags, dim3(4), dim3(256), 0, stream,
                     flags, NSTRIP * NCHUNK);

  dim3 grid(NSTRIP, NCHUNK);   // x (strips) fastest -> chunk-0 blocks first
  hipLaunchKernelGGL(cp1d_scan_kernel, grid, dim3(BLOCK),
                     SMEM_FLOATS * sizeof(float), stream,
                     x, t, s, b, y, carry, flags);
}